// CRFLoss_sf_34162169872846
// MI455X (gfx1250) — compile-verified
//
#include <hip/hip_runtime.h>
#include <hip/hip_bf16.h>
#include <stdint.h>

// Problem constants (from the reference)
#define L_SEQ      512
#define BATCH      128
#define T_TAG      20
#define TT         400            // T*T
#define TILE_BYTES 1600           // TT * 4
#define START_TAG  17
#define END_TAG    18
#define DEPTH      16             // async pipeline depth (tiles in flight)

#define LOG2E 1.4426950408889634f
#define LN2   0.6931471805599453f

// CDNA5 async global->LDS copy (gfx1250). VDST = LDS byte address VGPR,
// VADDR = 64-bit global address VGPR pair, GV mode ("off" = no SADDR).
// IOFFSET is applied to BOTH the LDS and the global address (ISA 15.18.3),
// so one base address pair serves all four 512B chunks of a tile.
#define ASYNC_LOAD_B128(ldsoff, gaddr, OFF)                                    \
  asm volatile("global_load_async_to_lds_b128 %0, %1, off offset:" #OFF        \
               :: "v"(ldsoff), "v"(gaddr) : "memory")

// 4 async instructions per tile; DEPTH tiles in flight -> oldest tile complete
// when ASYNCcnt <= 4*(DEPTH-1) = 60 (async loads complete in order; 6-bit cnt).
#define WAIT_ASYNCCNT_STEADY() asm volatile("s_wait_asynccnt 60" ::: "memory")
#define WAIT_ASYNCCNT_DRAIN()  asm volatile("s_wait_asynccnt 0"  ::: "memory")

__global__ __launch_bounds__(32)
void crf_chain_kernel(const float* __restrict__ trellis,
                      const int* __restrict__ gold,
                      const unsigned char* __restrict__ mask,
                      float* __restrict__ ws) {
  __shared__ float tile[DEPTH][TT];   // 16-deep ring of 20x20 tiles (raw trellis)
  __shared__ float pbuf[T_TAG];       // per-step broadcast buffer for partK[]

  const int b    = (int)blockIdx.x;   // one wave per batch element
  const int lane = (int)threadIdx.x;

  const unsigned long long gbase =
      (unsigned long long)(uintptr_t)trellis + (unsigned long long)b * TILE_BYTES;
  const unsigned lds_base = (unsigned)(uintptr_t)(&tile[0][0]); // low32 = LDS offset

  // Issue the 4 async b128 copies covering one 1600-byte tile.
  // Chunks 0..2 are full-wave (32 lanes x 16B); chunk 3 covers the last 64B.
  auto issue_tile = [&](int l) {
    const int slot = l & (DEPTH - 1);
    const unsigned long long g =
        gbase + (unsigned long long)l * (unsigned long long)(BATCH * TILE_BYTES)
              + (unsigned)(lane * 16);
    const unsigned lo = lds_base + (unsigned)(slot * TILE_BYTES) + (unsigned)(lane * 16);
    ASYNC_LOAD_B128(lo, g, 0);
    ASYNC_LOAD_B128(lo, g, 512);
    ASYNC_LOAD_B128(lo, g, 1024);
    if (lane < 4) { ASYNC_LOAD_B128(lo, g, 1536); }
  };

  // Prologue: fill the pipeline (HW stalls at the 6-bit ASYNCcnt cap; benign).
#pragma unroll 1
  for (int p = 0; p < DEPTH; ++p) issue_tile(p);

  float partK[T_TAG] = {};  // partition vector scaled by log2(e), replicated per lane
  float gacc = 0.0f;        // gold-energy partial (lane 20), in nats

  // Scalar side-data for step 0 (wave-uniform).
  int mon  = (int)mask[b];
  int gidx = gold[b];

  // ---- Peeled step l = 0: partition0 = trellis[0,b,START,:] (no cost) ----
  WAIT_ASYNCCNT_STEADY();
  if (lane == 20) {
    const float gv = tile[0][gidx];
    gacc += mon ? gv : 0.0f;
  }
  if (lane < T_TAG) pbuf[lane] = tile[0][START_TAG * T_TAG + lane] * LOG2E;
  __syncthreads();
#pragma unroll
  for (int i = 0; i < T_TAG; ++i) partK[i] = pbuf[i];
  __syncthreads();
  issue_tile(DEPTH);                 // keep the ring full
  mon  = (int)mask[BATCH + b];       // prefetch scalars for l = 1
  gidx = gold[BATCH + b];

  // ---- Steady loop l = 1 .. 511 ----
#pragma unroll 1
  for (int l = 1; l < L_SEQ; ++l) {
    if (l + DEPTH <= L_SEQ) { WAIT_ASYNCCNT_STEADY(); }
    else                    { WAIT_ASYNCCNT_DRAIN();  }

    const int slot = l & (DEPTH - 1);

    // Gold energy gather: raw trellis value straight from the resident tile.
    if (lane == 20) {
      const float gv = tile[slot][gidx];
      gacc += mon ? gv : 0.0f;
    }

    if (lane < T_TAG) {
      // base-2 domain: curK_j = max_i(v) + log2(sum 2^(v-max)) + cost_j*log2e
      // with v_i = trellis_ij*log2e + partK_i  (single v_fma per element)
      float vals[T_TAG];
      float m = -__builtin_inff();
#pragma unroll
      for (int i = 0; i < T_TAG; ++i) {
        const float v = __builtin_fmaf(tile[slot][i * T_TAG + lane], LOG2E, partK[i]);
        vals[i] = v;
        m = fmaxf(m, v);
      }
      float s = 0.0f;
#pragma unroll
      for (int i = 0; i < T_TAG; ++i) s += __builtin_amdgcn_exp2f(vals[i] - m);
      const float costK = (lane < 4) ? LOG2E : 0.0f;   // CHEM_COLS = {0,1,2,3}
      const float curK  = m + __builtin_amdgcn_logf(s) + costK;
      pbuf[lane] = mon ? curK : partK[lane];           // masked update
    }

    __syncthreads();          // single-wave block: effectively free
#pragma unroll
    for (int i = 0; i < T_TAG; ++i) partK[i] = pbuf[i]; // re-replicate into regs
    __syncthreads();

    // All DS reads of this slot were consumed above -> safe to reuse the slot.
    const int nl = l + DEPTH;
    if (nl < L_SEQ) issue_tile(nl);

    // Software-pipeline the wave-uniform scalar loads for the next step so the
    // SMEM latency overlaps the asynccnt wait + compute of this step.
    const int ln = l + 1;
    if (ln < L_SEQ) {
      const size_t lb = (size_t)ln * BATCH + (size_t)b;
      mon  = (int)mask[lb];
      gidx = gold[lb];
    }
  }

  if (lane == END_TAG) ws[b]         = partK[END_TAG] * LN2; // back to nats
  if (lane == 20)      ws[BATCH + b] = gacc;                 // per-b gold energy
}

// Deterministic final reduction: loss = (sum_b part18 - sum_b gold_b) / BATCH
__global__ __launch_bounds__(BATCH)
void crf_reduce_kernel(const float* __restrict__ ws, float* __restrict__ out) {
  __shared__ float sp[BATCH];
  const int t = (int)threadIdx.x;
  sp[t] = ws[t] - ws[BATCH + t];
  __syncthreads();
#pragma unroll
  for (int s = BATCH / 2; s > 0; s >>= 1) {
    if (t < s) sp[t] += sp[t + s];
    __syncthreads();
  }
  if (t == 0) out[0] = sp[0] / (float)BATCH;
}

extern "C" void kernel_launch(void* const* d_in, const int* in_sizes, int n_in,
                              void* d_out, int out_size, void* d_ws, size_t ws_size,
                              hipStream_t stream) {
  (void)in_sizes; (void)n_in; (void)out_size; (void)ws_size;
  const float*         trellis = (const float*)d_in[0];         // (512,128,20,20) f32
  const int*           gold    = (const int*)d_in[1];           // (512,128,1) int
  const unsigned char* mask    = (const unsigned char*)d_in[2]; // (512,128) bool
  // d_in[3] (corpus_mask) is unused by the math.
  float* ws  = (float*)d_ws;   // 256 floats: [0..127]=part18, [128..255]=gold
  float* out = (float*)d_out;  // 1 float

  crf_chain_kernel<<<dim3(BATCH), dim3(32), 0, stream>>>(trellis, gold, mask, ws);
  crf_reduce_kernel<<<dim3(1), dim3(BATCH), 0, stream>>>(ws, out);
}